// VectorQuantizerEMA_21303037788254
// MI455X (gfx1250) — compile-verified
//
#include <hip/hip_runtime.h>
#include <math.h>

typedef float v2f __attribute__((ext_vector_type(2)));
typedef float v8f __attribute__((ext_vector_type(8)));

#define EMB_D   256
#define MU_F    0.01f
#define EPS_F   1e-5f

// ---------------------------------------------------------------------------
// Utility: zero a float region
// ---------------------------------------------------------------------------
__global__ void vq_zero(float* __restrict__ p, int n) {
    int i = blockIdx.x * blockDim.x + threadIdx.x;
    if (i < n) p[i] = 0.0f;
}

__device__ __forceinline__ float wave_sum32(float v) {
#pragma unroll
    for (int off = 16; off > 0; off >>= 1)
        v += __shfl_xor(v, off, 32);
    return v;
}

// ---------------------------------------------------------------------------
// ||e_k||^2 for each code. One wave per code, 8 floats per lane.
// ---------------------------------------------------------------------------
__global__ __launch_bounds__(256) void vq_enorm(const float* __restrict__ E,
                                                float* __restrict__ enorm) {
    const int lane = threadIdx.x & 31;
    const int wave = threadIdx.x >> 5;
    const int k = blockIdx.x * 8 + wave;
    const float* e = E + (size_t)k * EMB_D;
    float s = 0.0f;
#pragma unroll
    for (int i = 0; i < 2; ++i) {
        const int col = lane * 4 + i * 128;
        float4 v = *(const float4*)(e + col);
        s += v.x * v.x + v.y * v.y + v.z * v.z + v.w * v.w;
    }
    s = wave_sum32(s);
    if (lane == 0) enorm[k] = s;
}

// ---------------------------------------------------------------------------
// Argmin over codes via V_WMMA_F32_16X16X4_F32.
// Each wave owns 16 tokens; the 16x256 A tile is register-resident
// (64 float2 per lane). Loop over 64 code tiles of 16 codes each;
// dist = ||e||^2 - 2 * (x . e)   (||x||^2 constant per token, dropped).
// Block = 256 threads = 8 waves = 128 tokens.
// ---------------------------------------------------------------------------
__global__ __launch_bounds__(256) void vq_argmin(const float* __restrict__ X,
                                                 const float* __restrict__ E,
                                                 const float* __restrict__ enorm,
                                                 float* __restrict__ idx_out,
                                                 int K) {
    const int lane = threadIdx.x & 31;
    const int wave = threadIdx.x >> 5;
    const int r = lane & 15;   // token row (lanes 0-15) / N index for B
    const int h = lane >> 4;   // half select: k-pair offset
    const int t0 = blockIdx.x * 128 + wave * 16;

    // A-matrix 16x4 layout: lane<16 -> M=r, K={4c,4c+1}; lane>=16 -> M=r, K={4c+2,4c+3}
    v2f a[64];
    const float* xrow = X + (size_t)(t0 + r) * EMB_D + 2 * h;
#pragma unroll
    for (int c = 0; c < 64; ++c)
        a[c] = *(const v2f*)(xrow + 4 * c);

    float bestD[8];
    int   bestI[8];
#pragma unroll
    for (int j = 0; j < 8; ++j) { bestD[j] = 3.4e38f; bestI[j] = 0; }

    for (int n0 = 0; n0 < K; n0 += 16) {
        // B-matrix 4x16 layout mirrors A on the N index
        const float* brow = E + (size_t)(n0 + r) * EMB_D + 2 * h;
        const float  en   = enorm[n0 + r];

        v8f acc = {0.f, 0.f, 0.f, 0.f, 0.f, 0.f, 0.f, 0.f};
#pragma unroll
        for (int c = 0; c < 64; ++c) {
            v2f b = *(const v2f*)(brow + 4 * c);
            acc = __builtin_amdgcn_wmma_f32_16x16x4_f32(
                false, a[c], false, b, (short)0, acc, false, false);
        }

        // C/D layout: VGPR j, lanes 0-15 -> M=j; lanes 16-31 -> M=8+j; N=lane%16
#pragma unroll
        for (int j = 0; j < 8; ++j) {
            float dist = en - 2.0f * acc[j];
            if (dist < bestD[j]) { bestD[j] = dist; bestI[j] = n0 + r; }
        }
    }

    // Reduce over the 16 codes (N=lane%16) within each 16-lane half.
#pragma unroll
    for (int j = 0; j < 8; ++j) {
        float d = bestD[j];
        int   i = bestI[j];
#pragma unroll
        for (int off = 1; off < 16; off <<= 1) {
            float od = __shfl_xor(d, off, 16);
            int   oi = __shfl_xor(i, off, 16);
            if (od < d || (od == d && oi < i)) { d = od; i = oi; }
        }
        if (r == 0)  // lane 0 -> tokens m=0..7, lane 16 -> tokens m=8..15
            idx_out[t0 + h * 8 + j] = (float)i;
    }
}

// ---------------------------------------------------------------------------
// Gather quantized rows, commitment-loss partial sum, counts, MU*dw scatter.
// One wave per token.
// ---------------------------------------------------------------------------
__global__ __launch_bounds__(256) void vq_gather_scatter(
    const float* __restrict__ X, const float* __restrict__ E,
    const float* __restrict__ idx_f,
    float* __restrict__ q_out, float* __restrict__ dw_acc,  // new_ema_w slot
    float* __restrict__ counts,                             // new_cluster slot
    float* __restrict__ loss_acc) {
    const int lane = threadIdx.x & 31;
    const int wave = threadIdx.x >> 5;
    const int t = blockIdx.x * 8 + wave;

    const int idx = (int)idx_f[t];
    const float* x = X + (size_t)t * EMB_D;
    const float* e = E + (size_t)idx * EMB_D;
    float* q = q_out + (size_t)t * EMB_D;
    float* dw = dw_acc + (size_t)idx * EMB_D;

    float s = 0.0f;
#pragma unroll
    for (int i = 0; i < 2; ++i) {
        const int col = lane * 4 + i * 128;
        float4 xv = *(const float4*)(x + col);
        float4 ev = *(const float4*)(e + col);
        *(float4*)(q + col) = ev;
        float dx = ev.x - xv.x, dy = ev.y - xv.y, dz = ev.z - xv.z, dw4 = ev.w - xv.w;
        s += dx * dx + dy * dy + dz * dz + dw4 * dw4;
        unsafeAtomicAdd(dw + col + 0, MU_F * xv.x);
        unsafeAtomicAdd(dw + col + 1, MU_F * xv.y);
        unsafeAtomicAdd(dw + col + 2, MU_F * xv.z);
        unsafeAtomicAdd(dw + col + 3, MU_F * xv.w);
    }
    s = wave_sum32(s);
    if (lane == 0) {
        unsafeAtomicAdd(loss_acc, s);
        unsafeAtomicAdd(&counts[idx], 1.0f);
    }
}

// ---------------------------------------------------------------------------
// EMA stage 1: pre-cluster, global n, entropy accumulation. One thread/code.
// ---------------------------------------------------------------------------
__global__ __launch_bounds__(256) void vq_c1(const float* __restrict__ counts,
                                             const float* __restrict__ ema_cs,
                                             float* __restrict__ pre_buf,
                                             float* __restrict__ n_acc,
                                             float* __restrict__ plex_acc,
                                             int N) {
    const int k = blockIdx.x * blockDim.x + threadIdx.x;
    const float c = counts[k];
    const float p = c / (float)N;
    unsafeAtomicAdd(plex_acc, p * logf(p + 1e-10f));
    const float pre = ema_cs[k] * (1.0f - MU_F) + MU_F * c;
    pre_buf[k] = pre;
    unsafeAtomicAdd(n_acc, pre);
}

// ---------------------------------------------------------------------------
// EMA stage 2: finalize new_cluster / new_ema_w / new_embedding / scalars.
// Grid = K blocks, block = 256 (one thread per (k, d)).
// ---------------------------------------------------------------------------
__global__ __launch_bounds__(256) void vq_c2(const float* __restrict__ pre_buf,
                                             const float* __restrict__ ema_w,
                                             float* __restrict__ new_ema_w,  // holds MU*dw
                                             float* __restrict__ new_emb,
                                             float* __restrict__ new_cluster,
                                             const float* __restrict__ n_acc,
                                             const float* __restrict__ loss_acc,
                                             const float* __restrict__ plex_acc,
                                             float* __restrict__ out_loss,
                                             float* __restrict__ out_perp,
                                             int N, int K) {
    const int k = blockIdx.x;
    const int d = threadIdx.x;
    const size_t gid = (size_t)k * EMB_D + d;

    const float n = *n_acc;
    const float pre = pre_buf[k];
    const float newc = (pre + EPS_F) / (n + (float)K * EPS_F) * n;

    const float nw = ema_w[gid] * (1.0f - MU_F) + new_ema_w[gid];
    new_ema_w[gid] = nw;
    new_emb[gid] = nw / newc;

    if (d == 0) new_cluster[k] = newc;
    if (k == 0 && d == 0) {
        *out_loss = 0.25f * (*loss_acc) / ((float)N * (float)EMB_D);
        *out_perp = expf(-(*plex_acc));
    }
}

// ---------------------------------------------------------------------------
extern "C" void kernel_launch(void* const* d_in, const int* in_sizes, int n_in,
                              void* d_out, int out_size, void* d_ws, size_t ws_size,
                              hipStream_t stream) {
    const float* X    = (const float*)d_in[0];  // [N, 256]
    const float* E    = (const float*)d_in[1];  // [K, 256]
    const float* EMAW = (const float*)d_in[2];  // [K, 256]
    const float* EMAC = (const float*)d_in[3];  // [K]

    const int N = in_sizes[0] / EMB_D;
    const int K = in_sizes[1] / EMB_D;

    // d_out layout (flat f32, reference return order)
    float* out = (float*)d_out;
    size_t o = 0;
    float* out_q    = out + o; o += (size_t)N * EMB_D;  // quantized
    float* out_loss = out + o; o += 1;                  // loss
    float* out_idx  = out + o; o += (size_t)N;          // encoding_indices
    float* out_perp = out + o; o += 1;                  // perplexity
    float* out_nc   = out + o; o += (size_t)K;          // new_cluster (counts accum first)
    float* out_nw   = out + o; o += (size_t)K * EMB_D;  // new_ema_w (MU*dw accum first)
    float* out_ne   = out + o; o += (size_t)K * EMB_D;  // new_embedding

    // workspace: enorm[K], pre[K], 3 scalar accumulators
    float* ws       = (float*)d_ws;
    float* enorm    = ws;
    float* pre_buf  = ws + K;
    float* loss_acc = ws + 2 * K;      // [0]=loss, [1]=n, [2]=plex
    float* n_acc    = loss_acc + 1;
    float* plex_acc = loss_acc + 2;

    // Re-zero accumulator regions every call (harness does not re-poison).
    vq_zero<<<(K * EMB_D + 255) / 256, 256, 0, stream>>>(out_nw, K * EMB_D);
    vq_zero<<<(K + 255) / 256, 256, 0, stream>>>(out_nc, K);
    vq_zero<<<1, 32, 0, stream>>>(loss_acc, 3);

    vq_enorm<<<K / 8, 256, 0, stream>>>(E, enorm);
    vq_argmin<<<N / 128, 256, 0, stream>>>(X, E, enorm, out_idx, K);
    vq_gather_scatter<<<N / 8, 256, 0, stream>>>(X, E, out_idx, out_q, out_nw,
                                                 out_nc, loss_acc);
    vq_c1<<<K / 256, 256, 0, stream>>>(out_nc, EMAC, pre_buf, n_acc, plex_acc, N);
    vq_c2<<<K, 256, 0, stream>>>(pre_buf, EMAW, out_nw, out_ne, out_nc,
                                 n_acc, loss_acc, plex_acc, out_loss, out_perp, N, K);
}